// Intention_15083925144155
// MI455X (gfx1250) — compile-verified
//
#include <hip/hip_runtime.h>

typedef float v2f __attribute__((ext_vector_type(2)));
typedef float v8f __attribute__((ext_vector_type(8)));

#define WMMA_F32(a, b, c) \
  __builtin_amdgcn_wmma_f32_16x16x4_f32(false, (a), false, (b), (short)0, (c), false, false)

#if defined(__has_builtin)
#if __has_builtin(__builtin_amdgcn_global_load_async_to_lds_b128)
#define HAVE_ASYNC_LDS 1
#endif
#if __has_builtin(__builtin_amdgcn_s_wait_asynccnt)
#define HAVE_WAIT_ASYNC_BUILTIN 1
#endif
#endif

__device__ __forceinline__ void wait_asynccnt0() {
#if defined(HAVE_WAIT_ASYNC_BUILTIN)
  __builtin_amdgcn_s_wait_asynccnt(0);
#else
  asm volatile("s_wait_asynccnt 0x0" ::: "memory");
#endif
}

constexpr int Bc = 4, Nc = 4096, Dc = 1024, Hc = 16, HDc = 64;
constexpr int Mc = Bc * Nc;  // 16384 rows when flattened

// LDS row stride for 64x16 tiles: 20 floats = 80 bytes -> every row start is
// 16B aligned (for b128 async stores) and gcd(20,64)=4 keeps 16-lane
// transposed reads on 16 distinct banks.
constexpr int LSTR = 20;

// ---------------------------------------------------------------------------
// C[M,Nn] = X[M,K] @ W[Nn,K]^T  (+ bias[Nn]) (+ resid[M,Nn])
// block: 128 threads = 4 waves; C tile 64x64; wave w owns rows w*16..w*16+15,
// all four 16-col tiles. K staged in LDS in chunks of 16 (4 WMMA k=4 steps),
// double-buffered via GLOBAL_LOAD_ASYNC_TO_LDS_B128 when available.
// ---------------------------------------------------------------------------
__global__ __launch_bounds__(128) void gemm_xwt(
    const float* __restrict__ X, const float* __restrict__ W,
    float* __restrict__ C, const float* __restrict__ bias,
    const float* __restrict__ resid, int M, int Nn, int K) {
  __shared__ float Xs[2][64][LSTR];  // [buf][row(m)][k]
  __shared__ float Wl[2][64][LSTR];  // [buf][row(n)][k]  (B read transposed)
  const int t = threadIdx.x;
  const int wave = t >> 5, lane = t & 31, half = lane >> 4, l15 = lane & 15;
  const int n0 = blockIdx.x * 64, m0 = blockIdx.y * 64;
  const int lrow = t >> 1, kg = (t & 1) * 8;  // each thread: 8 floats of one row

  v8f acc[4] = {};

  const float* gx_row = X + (size_t)(m0 + lrow) * K + kg;
  const float* gw_row = W + (size_t)(n0 + lrow) * K + kg;

  auto compute = [&](int buf) {
#pragma unroll
    for (int ks = 0; ks < 16; ks += 4) {
      v2f a;
      a.x = Xs[buf][wave * 16 + l15][ks + 2 * half];
      a.y = Xs[buf][wave * 16 + l15][ks + 2 * half + 1];
#pragma unroll
      for (int nt = 0; nt < 4; ++nt) {
        v2f b;
        b.x = Wl[buf][nt * 16 + l15][ks + 2 * half];
        b.y = Wl[buf][nt * 16 + l15][ks + 2 * half + 1];
        acc[nt] = WMMA_F32(a, b, acc[nt]);
      }
    }
  };

#if defined(HAVE_ASYNC_LDS)
  typedef int v4i __attribute__((vector_size(16)));
  typedef __attribute__((address_space(1))) v4i gv4;  // global int4
  typedef __attribute__((address_space(3))) v4i lv4;  // LDS int4

  auto issue = [&](int buf, int k0) {
    __builtin_amdgcn_global_load_async_to_lds_b128(
        (gv4*)(gx_row + k0), (lv4*)&Xs[buf][lrow][kg], 0, 0);
    __builtin_amdgcn_global_load_async_to_lds_b128(
        (gv4*)(gx_row + k0 + 4), (lv4*)&Xs[buf][lrow][kg + 4], 0, 0);
    __builtin_amdgcn_global_load_async_to_lds_b128(
        (gv4*)(gw_row + k0), (lv4*)&Wl[buf][lrow][kg], 0, 0);
    __builtin_amdgcn_global_load_async_to_lds_b128(
        (gv4*)(gw_row + k0 + 4), (lv4*)&Wl[buf][lrow][kg + 4], 0, 0);
  };

  issue(0, 0);
  wait_asynccnt0();
  __syncthreads();
  for (int k0 = 0; k0 < K; k0 += 16) {
    const int buf = (k0 >> 4) & 1;
    const bool more = (k0 + 16) < K;
    if (more) issue(buf ^ 1, k0 + 16);  // overlap next chunk with WMMAs
    compute(buf);
    if (more) {
      wait_asynccnt0();
      __syncthreads();  // also orders LDS reads of `buf` before next overwrite
    }
  }
#else
  for (int k0 = 0; k0 < K; k0 += 16) {
    const float4* xp = reinterpret_cast<const float4*>(gx_row + k0);
    const float4* wp = reinterpret_cast<const float4*>(gw_row + k0);
    float4 xa = xp[0], xb = xp[1];
    float4 wa = wp[0], wb = wp[1];
    Xs[0][lrow][kg + 0] = xa.x; Xs[0][lrow][kg + 1] = xa.y;
    Xs[0][lrow][kg + 2] = xa.z; Xs[0][lrow][kg + 3] = xa.w;
    Xs[0][lrow][kg + 4] = xb.x; Xs[0][lrow][kg + 5] = xb.y;
    Xs[0][lrow][kg + 6] = xb.z; Xs[0][lrow][kg + 7] = xb.w;
    Wl[0][lrow][kg + 0] = wa.x; Wl[0][lrow][kg + 1] = wa.y;
    Wl[0][lrow][kg + 2] = wa.z; Wl[0][lrow][kg + 3] = wa.w;
    Wl[0][lrow][kg + 4] = wb.x; Wl[0][lrow][kg + 5] = wb.y;
    Wl[0][lrow][kg + 6] = wb.z; Wl[0][lrow][kg + 7] = wb.w;
    __syncthreads();
    compute(0);
    __syncthreads();
  }
#endif

#pragma unroll
  for (int nt = 0; nt < 4; ++nt) {
    const int colg = n0 + nt * 16 + l15;
#pragma unroll
    for (int r = 0; r < 8; ++r) {
      const int rowg = m0 + wave * 16 + r + 8 * half;
      float v = acc[nt][r];
      if (bias) v += bias[colg];
      if (resid) v += resid[(size_t)rowg * Nn + colg];
      C[(size_t)rowg * Nn + colg] = v;
    }
  }
}

// ---------------------------------------------------------------------------
// Per (b,h): kk = K^T K + alpha*I, ktv = K^T V   (both 64x64, sum over N=4096)
// 256 threads = 8 waves; waves 0-3 -> kk row strips, waves 4-7 -> ktv strips.
// ---------------------------------------------------------------------------
__global__ __launch_bounds__(256) void kk_ktv_kernel(
    const float* __restrict__ Kx, const float* __restrict__ Vx,
    const float* __restrict__ alpha, float* __restrict__ kkb,
    float* __restrict__ ktvb) {
  __shared__ float Kt[64][65];  // [n][d]
  __shared__ float Vt[64][65];
  const int bh = blockIdx.x, b = bh >> 4, h = bh & 15;
  const int t = threadIdx.x;
  const int wave = t >> 5, lane = t & 31, half = lane >> 4, l15 = lane & 15;
  const int mstrip = (wave & 3) * 16;
  const bool isK = (wave < 4);  // wave-uniform: EXEC stays full for WMMA
  const int lrow = t >> 2, cg = (t & 3) * 16;

  v8f acc[4] = {};

  for (int nb = 0; nb < Nc; nb += 64) {
    const size_t gbase = ((size_t)(b * Nc + nb + lrow)) * Dc + h * HDc + cg;
    const float4* kp = reinterpret_cast<const float4*>(Kx + gbase);
    const float4* vp = reinterpret_cast<const float4*>(Vx + gbase);
#pragma unroll
    for (int i = 0; i < 4; ++i) {
      float4 kv = kp[i], vv = vp[i];
      Kt[lrow][cg + 4 * i + 0] = kv.x; Kt[lrow][cg + 4 * i + 1] = kv.y;
      Kt[lrow][cg + 4 * i + 2] = kv.z; Kt[lrow][cg + 4 * i + 3] = kv.w;
      Vt[lrow][cg + 4 * i + 0] = vv.x; Vt[lrow][cg + 4 * i + 1] = vv.y;
      Vt[lrow][cg + 4 * i + 2] = vv.z; Vt[lrow][cg + 4 * i + 3] = vv.w;
    }
    __syncthreads();

#pragma unroll 4
    for (int kb = 0; kb < 64; kb += 4) {
      v2f a;  // A = K^T : A[d][n] = Kt[n][d]
      a.x = Kt[kb + 2 * half][mstrip + l15];
      a.y = Kt[kb + 2 * half + 1][mstrip + l15];
#pragma unroll
      for (int nt = 0; nt < 4; ++nt) {
        v2f bvec;
        if (isK) {
          bvec.x = Kt[kb + 2 * half][nt * 16 + l15];
          bvec.y = Kt[kb + 2 * half + 1][nt * 16 + l15];
        } else {
          bvec.x = Vt[kb + 2 * half][nt * 16 + l15];
          bvec.y = Vt[kb + 2 * half + 1][nt * 16 + l15];
        }
        acc[nt] = WMMA_F32(a, bvec, acc[nt]);
      }
    }
    __syncthreads();
  }

  const float al = alpha[0];
  float* dst = isK ? kkb : ktvb;
  const size_t base = (size_t)bh * HDc * HDc;
#pragma unroll
  for (int nt = 0; nt < 4; ++nt) {
    const int colg = nt * 16 + l15;
#pragma unroll
    for (int r = 0; r < 8; ++r) {
      const int rowl = mstrip + r + 8 * half;
      float v = acc[nt][r];
      if (isK && rowl == colg) v += al;
      dst[base + (size_t)rowl * HDc + colg] = v;
    }
  }
}

// ---------------------------------------------------------------------------
// Per (b,h): solve kk * X = ktv via Gauss-Jordan (kk is SPD -> no pivoting),
// then softmax over rows (axis=-2) with temperature. 128 threads, one column
// of the augmented [kk | ktv] per thread.
// ---------------------------------------------------------------------------
__global__ __launch_bounds__(128) void solve_softmax(
    const float* __restrict__ kkb, const float* __restrict__ ktvb,
    const float* __restrict__ temp, float* __restrict__ attnb) {
  __shared__ float aug[64][130];
  __shared__ float colbuf[64];
  const int bh = blockIdx.x, t = threadIdx.x;
  const size_t base = (size_t)bh * HDc * HDc;

  for (int idx = t; idx < 64 * 128; idx += 128) {
    const int r = idx >> 7, c = idx & 127;
    aug[r][c] = (c < 64) ? kkb[base + r * 64 + c] : ktvb[base + r * 64 + (c - 64)];
  }
  __syncthreads();

  const int col = t;
  for (int i = 0; i < 64; ++i) {
    if (t < 64) colbuf[t] = aug[t][i];  // snapshot pivot column
    __syncthreads();
    const float ric = aug[i][col] / colbuf[i];
    aug[i][col] = ric;
#pragma unroll 4
    for (int j = 0; j < 64; ++j) {
      if (j != i) aug[j][col] -= colbuf[j] * ric;
    }
    __syncthreads();
  }

  // right half now holds kk_inv @ ktv; softmax over d (rows) per column e
  const float invT = 1.0f / temp[0];
  if (t < 64) {
    const int e = t;
    float mx = -3.0e38f;
    for (int d = 0; d < 64; ++d) mx = fmaxf(mx, aug[d][64 + e] * invT);
    float s = 0.0f;
    for (int d = 0; d < 64; ++d) {
      const float v = expf(aug[d][64 + e] * invT - mx);
      aug[d][e] = v;  // reuse left half as scratch
      s += v;
    }
    const float inv_s = 1.0f / s;
    for (int d = 0; d < 64; ++d) attnb[base + d * 64 + e] = aug[d][e] * inv_s;
  }
}

// ---------------------------------------------------------------------------
// out[b,n,h*64+e] = sum_d q[b,n,h*64+d] * attn[bh][d][e]
// grid: (N/64, B*H); 128 threads = 4 waves, 64x64 tile, K = 64.
// ---------------------------------------------------------------------------
__global__ __launch_bounds__(128) void qattn_kernel(
    const float* __restrict__ Q, const float* __restrict__ attnb,
    float* __restrict__ Out) {
  __shared__ float At[64][65];  // [d][e]
  __shared__ float Qs[64][65];  // [n][d]
  const int n0 = blockIdx.x * 64;
  const int bh = blockIdx.y, b = bh >> 4, h = bh & 15;
  const int t = threadIdx.x;
  const int wave = t >> 5, lane = t & 31, half = lane >> 4, l15 = lane & 15;
  const size_t abase = (size_t)bh * HDc * HDc;

  for (int idx = t; idx < 64 * 64; idx += 128)
    At[idx >> 6][idx & 63] = attnb[abase + idx];

  const int lrow = t >> 1, cg = (t & 1) * 32;
  const float4* qp = reinterpret_cast<const float4*>(
      Q + ((size_t)(b * Nc + n0 + lrow)) * Dc + h * HDc + cg);
#pragma unroll
  for (int i = 0; i < 8; ++i) {
    float4 qv = qp[i];
    Qs[lrow][cg + 4 * i + 0] = qv.x; Qs[lrow][cg + 4 * i + 1] = qv.y;
    Qs[lrow][cg + 4 * i + 2] = qv.z; Qs[lrow][cg + 4 * i + 3] = qv.w;
  }
  __syncthreads();

  v8f acc[4] = {};
  const int mstrip = wave * 16;
#pragma unroll 4
  for (int kb = 0; kb < 64; kb += 4) {
    v2f a;
    a.x = Qs[mstrip + l15][kb + 2 * half];
    a.y = Qs[mstrip + l15][kb + 2 * half + 1];
#pragma unroll
    for (int nt = 0; nt < 4; ++nt) {
      v2f bvec;
      bvec.x = At[kb + 2 * half][nt * 16 + l15];
      bvec.y = At[kb + 2 * half + 1][nt * 16 + l15];
      acc[nt] = WMMA_F32(a, bvec, acc[nt]);
    }
  }

#pragma unroll
  for (int nt = 0; nt < 4; ++nt) {
    const int colg = nt * 16 + l15;
#pragma unroll
    for (int r = 0; r < 8; ++r) {
      const int rowl = mstrip + r + 8 * half;
      Out[((size_t)(b * Nc + n0 + rowl)) * Dc + h * HDc + colg] = acc[nt][r];
    }
  }
}

// ---------------------------------------------------------------------------
// In-place LayerNorm over last dim (D=1024). One block per row, 256 threads.
// ---------------------------------------------------------------------------
__global__ __launch_bounds__(256) void ln_kernel(float* __restrict__ Y,
                                                 const float* __restrict__ gamma,
                                                 const float* __restrict__ beta) {
  __shared__ float s1[256], s2[256];
  const size_t base = (size_t)blockIdx.x * Dc;
  const int t = threadIdx.x;
  float v[4];
  float sum = 0.f, sq = 0.f;
#pragma unroll
  for (int i = 0; i < 4; ++i) {
    v[i] = Y[base + t + 256 * i];
    sum += v[i];
    sq += v[i] * v[i];
  }
  s1[t] = sum; s2[t] = sq;
  __syncthreads();
  for (int off = 128; off > 0; off >>= 1) {
    if (t < off) { s1[t] += s1[t + off]; s2[t] += s2[t + off]; }
    __syncthreads();
  }
  const float mu = s1[0] * (1.0f / Dc);
  const float var = s2[0] * (1.0f / Dc) - mu * mu;
  const float rs = rsqrtf(var + 1e-5f);
#pragma unroll
  for (int i = 0; i < 4; ++i) {
    const int c = t + 256 * i;
    Y[base + c] = (v[i] - mu) * rs * gamma[c] + beta[c];
  }
}

// ---------------------------------------------------------------------------
extern "C" void kernel_launch(void* const* d_in, const int* in_sizes, int n_in,
                              void* d_out, int out_size, void* d_ws,
                              size_t ws_size, hipStream_t stream) {
  const float* x     = (const float*)d_in[0];
  const float* Wq    = (const float*)d_in[1];
  const float* Wk    = (const float*)d_in[2];
  const float* Wv    = (const float*)d_in[3];
  const float* Wo    = (const float*)d_in[4];
  const float* bo    = (const float*)d_in[5];
  const float* alpha = (const float*)d_in[6];
  const float* temp  = (const float*)d_in[7];
  const float* gamma = (const float*)d_in[8];
  const float* beta  = (const float*)d_in[9];
  float* out = (float*)d_out;
  float* ws  = (float*)d_ws;

  size_t off = 0;
  float* qb = ws + off;  off += (size_t)Mc * Dc;   // 64 MB
  float* kb = ws + off;  off += (size_t)Mc * Dc;   // 64 MB (reused as attn_out)
  float* vb = ws + off;  off += (size_t)Mc * Dc;   // 64 MB
  float* kkb   = ws + off; off += (size_t)Bc * Hc * HDc * HDc;
  float* ktvb  = ws + off; off += (size_t)Bc * Hc * HDc * HDc;
  float* attnb = ws + off;

  const dim3 gg(Dc / 64, Mc / 64);
  gemm_xwt<<<gg, 128, 0, stream>>>(x, Wq, qb, nullptr, nullptr, Mc, Dc, Dc);
  gemm_xwt<<<gg, 128, 0, stream>>>(x, Wk, kb, nullptr, nullptr, Mc, Dc, Dc);
  gemm_xwt<<<gg, 128, 0, stream>>>(x, Wv, vb, nullptr, nullptr, Mc, Dc, Dc);
  kk_ktv_kernel<<<Bc * Hc, 256, 0, stream>>>(kb, vb, alpha, kkb, ktvb);
  solve_softmax<<<Bc * Hc, 128, 0, stream>>>(kkb, ktvb, temp, attnb);
  qattn_kernel<<<dim3(Nc / 64, Bc * Hc), 128, 0, stream>>>(qb, attnb, kb);
  gemm_xwt<<<gg, 128, 0, stream>>>(kb, Wo, out, bo, x, Mc, Dc, Dc);
  ln_kernel<<<Mc, 256, 0, stream>>>(out, gamma, beta);
}